// A3C_Model_22832046146023
// MI455X (gfx1250) — compile-verified
//
#include <hip/hip_runtime.h>
#include <hip/hip_bf16.h>
#include <math.h>

#define N_NODES     10000
#define N_EDGES     640000
#define STATE_DIM   128
#define ACTION_DIM  10000
#define CONCAT_DIM  10003
#define H1_DIM      60
#define H2_DIM      30

typedef float v2f __attribute__((ext_vector_type(2)));
typedef float v8f __attribute__((ext_vector_type(8)));

static __device__ __forceinline__ int gid() {
    return blockIdx.x * blockDim.x + threadIdx.x;
}

// ---------------- utility kernels ----------------

__global__ void fill_kernel(float* __restrict__ p, float v, int n) {
    int i = gid();
    if (i < n) p[i] = v;
}

__global__ void deg_kernel(const int* __restrict__ dst, float* __restrict__ deg) {
    int e = gid();
    if (e < N_EDGES) atomicAdd(&deg[dst[e]], 1.0f);
}

__global__ void dinv_kernel(float* __restrict__ deg) {
    int i = gid();
    if (i < N_NODES) {
        float d = deg[i];
        deg[i] = (d > 0.0f) ? (1.0f / sqrtf(d)) : 0.0f;
    }
}

__global__ void edgew_kernel(const int* __restrict__ src, const int* __restrict__ dst,
                             const float* __restrict__ dinv, float* __restrict__ w) {
    int e = gid();
    if (e < N_EDGES) w[e] = -dinv[src[e]] * dinv[dst[e]];
}

// Zero-pad weights [3,K,H] -> [3,K,HP] and bias [H] -> [HP]
__global__ void pad_wb_kernel(const float* __restrict__ W, const float* __restrict__ b,
                              float* __restrict__ Wp, float* __restrict__ bp,
                              int K, int H, int HP) {
    int idx = gid();
    int totalW = 3 * K * HP;
    if (idx < totalW) {
        int c  = idx % HP;
        int rk = idx / HP;                       // hop*K + k
        Wp[idx] = (c < H) ? W[rk * H + c] : 0.0f;
    } else if (idx < totalW + HP) {
        int c = idx - totalW;
        bp[c] = (c < H) ? b[c] : 0.0f;
    }
}

// y[dst, f] += w[e] * x[src, f] ; one thread per (edge, feature)
__global__ void spmv_kernel(const int* __restrict__ src, const int* __restrict__ dst,
                            const float* __restrict__ w, const float* __restrict__ x,
                            float* __restrict__ y, int F, int ld) {
    unsigned idx = (unsigned)gid();
    unsigned total = (unsigned)N_EDGES * (unsigned)F;
    if (idx >= total) return;
    int e = idx / F;
    int f = idx - e * F;
    atomicAdd(&y[dst[e] * ld + f], w[e] * x[src[e] * ld + f]);
}

// y = 2*y - x0  (elementwise)
__global__ void cheb_comb_kernel(float* __restrict__ y, const float* __restrict__ x0, int n) {
    int i = gid();
    if (i < n) y[i] = 2.0f * y[i] - x0[i];
}

// ---------------- fused ChebConv dense GEMM (WMMA f32 16x16x4) ----------------
// out[m, c] = tanh( sum_{hop} (T_hop @ Wp[hop])[m, c] + bp[c] )  for c < LDOUT
// Wp is zero-padded [3, K, HP]; grid = (M/16, HP/16), block = one wave32.
// No divergence before the WMMAs: EXEC stays all-ones as required.
template <int LDIN, int K, int HP, int LDOUT>
__global__ __launch_bounds__(32)
void cheb_gemm_wmma(const float* __restrict__ T0, const float* __restrict__ T1,
                    const float* __restrict__ T2, const float* __restrict__ Wp,
                    const float* __restrict__ bp, float* __restrict__ out) {
    const int mt   = blockIdx.x * 16;
    const int nt   = blockIdx.y * 16;
    const int lane = threadIdx.x;          // 0..31
    const int l15  = lane & 15;
    const int half = lane >> 4;            // 0 or 1
    const int col  = nt + l15;             // B / C column for this lane
    const int kq   = half * 2;             // K sub-offset within 16x4 A tile

    // Per-lane base pointers (32-bit offsets, all < 2^31)
    const int aoff = (mt + l15) * LDIN + kq;
    const float* __restrict__ A0 = T0 + aoff;
    const float* __restrict__ A1 = T1 + aoff;
    const float* __restrict__ A2 = T2 + aoff;
    const float* __restrict__ B0 = Wp + col + kq * HP;   // + hop*K*HP + k0*HP

    v8f acc = {0.f, 0.f, 0.f, 0.f, 0.f, 0.f, 0.f, 0.f};

    #pragma unroll
    for (int hop = 0; hop < 3; ++hop) {
        const float* __restrict__ X  = (hop == 0) ? A0 : ((hop == 1) ? A1 : A2);
        const float* __restrict__ Bh = B0 + hop * K * HP;
        #pragma unroll
        for (int k0 = 0; k0 < K; k0 += 4) {
            v2f a, b;
            // A 16x4 f32: lanes 0-15 hold K={k0,k0+1}, lanes 16-31 hold K={k0+2,k0+3}
            a.x = X[k0 + 0];
            a.y = X[k0 + 1];
            // B 4x16 f32: row = k0+kq+j striped HP apart, col striped across lanes
            b.x = Bh[(k0 + 0) * HP];
            b.y = Bh[(k0 + 1) * HP];
            acc = __builtin_amdgcn_wmma_f32_16x16x4_f32(
                /*neg_a=*/false, a, /*neg_b=*/false, b,
                /*c_mod=*/(short)0, acc, /*reuse_a=*/false, /*reuse_b=*/false);
        }
    }

    // C/D layout: VGPR r -> row mt + r + 8*half, column = nt + (lane&15)
    if (col < LDOUT) {
        const float bv = bp[col];
        const int rowBase = mt + half * 8;
        #pragma unroll
        for (int r = 0; r < 8; ++r) {
            out[(rowBase + r) * LDOUT + col] = tanhf(acc[r] + bv);
        }
    }
}

// ---------------- layer 3 (H=1) + state assembly ----------------
__global__ void layer3_state_kernel(const float* __restrict__ T0, const float* __restrict__ T1,
                                    const float* __restrict__ T2, const float* __restrict__ W3,
                                    const float* __restrict__ b3,
                                    const float* __restrict__ vcpu, const float* __restrict__ vbw,
                                    const float* __restrict__ vpend,
                                    float* __restrict__ state, int ldin) {
    int n = gid();
    if (n == 0) {
        state[N_NODES + 0] = vcpu[0];
        state[N_NODES + 1] = vbw[0];
        state[N_NODES + 2] = vpend[0];
    }
    if (n >= N_NODES) return;
    float s = b3[0];
    const float* rows[3] = {T0 + n * ldin, T1 + n * ldin, T2 + n * ldin};
    #pragma unroll
    for (int hop = 0; hop < 3; ++hop) {
        const float* x  = rows[hop];
        const float* wv = W3 + hop * H2_DIM;   // [3,30,1] -> stride 30
        #pragma unroll
        for (int k = 0; k < H2_DIM; ++k) s += x[k] * wv[k];
    }
    state[n] = tanhf(s);
}

// ---------------- actor/critic head: 400MB streaming matvec ----------------
// wave r (r<10000): out[r] = dot(state, Wa[r,:]) + ba[r]; wave 10000: critic.
__global__ void head_kernel(const float* __restrict__ state, const float* __restrict__ Wa,
                            const float* __restrict__ ba, const float* __restrict__ Wc,
                            const float* __restrict__ bc, float* __restrict__ out) {
    const int wave = (blockIdx.x * blockDim.x + threadIdx.x) >> 5;
    const int lane = threadIdx.x & 31;
    if (wave > ACTION_DIM) return;
    const float* __restrict__ row =
        (wave < ACTION_DIM) ? (Wa + (long long)wave * CONCAT_DIM) : Wc;
    float acc = 0.0f;
    for (int i = lane; i < CONCAT_DIM; i += 32) {
        if (i + 4096 < CONCAT_DIM)
            __builtin_prefetch((const void*)(row + i + 4096), 0, 0);
        acc += state[i] * row[i];
    }
    #pragma unroll
    for (int off = 16; off > 0; off >>= 1)
        acc += __shfl_down(acc, off, 32);
    if (lane == 0)
        out[wave] = acc + ((wave < ACTION_DIM) ? ba[wave] : bc[0]);
}

// ---------------- launch ----------------

static inline int cdiv(long long a, long long b) { return (int)((a + b - 1) / b); }

extern "C" void kernel_launch(void* const* d_in, const int* in_sizes, int n_in,
                              void* d_out, int out_size, void* d_ws, size_t ws_size,
                              hipStream_t stream) {
    const float* feat  = (const float*)d_in[0];     // [10000,128]
    const int*   eidx  = (const int*)  d_in[1];     // [2,640000]
    const float* vcpu  = (const float*)d_in[2];
    const float* vbw   = (const float*)d_in[3];
    const float* vpend = (const float*)d_in[4];
    const float* W1 = (const float*)d_in[5];  const float* b1 = (const float*)d_in[6];
    const float* W2 = (const float*)d_in[7];  const float* b2 = (const float*)d_in[8];
    const float* W3 = (const float*)d_in[9];  const float* b3 = (const float*)d_in[10];
    const float* Wa = (const float*)d_in[11]; const float* ba = (const float*)d_in[12];
    const float* Wc = (const float*)d_in[13]; const float* bc = (const float*)d_in[14];
    float* out = (float*)d_out;                     // 10001 floats: logits..., value

    const int* srcp = eidx;
    const int* dstp = eidx + N_EDGES;

    // workspace layout (floats)
    float* ws   = (float*)d_ws;
    float* deg  = ws;                                   // 10016 (deg -> dinv in place)
    float* ew   = deg + 10016;                          // 640000 edge weights
    float* T1   = ew + N_EDGES;                         // 10000*128
    float* T2   = T1 + N_NODES * STATE_DIM;             // 10000*128
    float* h1   = T2 + N_NODES * STATE_DIM;             // 10000*60
    float* h2   = h1 + N_NODES * H1_DIM;                // 10000*32 (ld 32, cols 30,31 zero)
    float* st   = h2 + N_NODES * 32;                    // 10016 state vector
    float* Wp1  = st + 10016;                           // 3*128*64
    float* bp1  = Wp1 + 3 * STATE_DIM * 64;             // 64
    float* Wp2  = bp1 + 64;                             // 3*60*32
    float* bp2  = Wp2 + 3 * H1_DIM * 32;                // 32

    const int B = 256;

    // ---- edge normalization ----
    fill_kernel <<<cdiv(N_NODES, B), B, 0, stream>>>(deg, 0.0f, N_NODES);
    deg_kernel  <<<cdiv(N_EDGES, B), B, 0, stream>>>(dstp, deg);
    dinv_kernel <<<cdiv(N_NODES, B), B, 0, stream>>>(deg);
    edgew_kernel<<<cdiv(N_EDGES, B), B, 0, stream>>>(srcp, dstp, deg, ew);

    // ---- pad weights for divergence-free WMMA B loads ----
    pad_wb_kernel<<<cdiv(3 * STATE_DIM * 64 + 64, B), B, 0, stream>>>(W1, b1, Wp1, bp1,
                                                                      STATE_DIM, H1_DIM, 64);
    pad_wb_kernel<<<cdiv(3 * H1_DIM * 32 + 32, B), B, 0, stream>>>(W2, b2, Wp2, bp2,
                                                                   H1_DIM, H2_DIM, 32);

    // ---- layer 1 : F=128, ld=128 ----
    {
        const int F = STATE_DIM, ld = STATE_DIM;
        const int nElem = N_NODES * ld;
        fill_kernel<<<cdiv(nElem, B), B, 0, stream>>>(T1, 0.0f, nElem);
        spmv_kernel<<<cdiv((long long)N_EDGES * F, B), B, 0, stream>>>(srcp, dstp, ew, feat, T1, F, ld);
        fill_kernel<<<cdiv(nElem, B), B, 0, stream>>>(T2, 0.0f, nElem);
        spmv_kernel<<<cdiv((long long)N_EDGES * F, B), B, 0, stream>>>(srcp, dstp, ew, T1, T2, F, ld);
        cheb_comb_kernel<<<cdiv(nElem, B), B, 0, stream>>>(T2, feat, nElem);
        dim3 g(N_NODES / 16, 64 / 16);   // 625 x 4 tiles
        cheb_gemm_wmma<STATE_DIM, STATE_DIM, 64, H1_DIM>
            <<<g, 32, 0, stream>>>(feat, T1, T2, Wp1, bp1, h1);
    }

    // ---- layer 2 : F=60, ld=60 ----
    {
        const int F = H1_DIM, ld = H1_DIM;
        const int nElem = N_NODES * ld;
        fill_kernel<<<cdiv(nElem, B), B, 0, stream>>>(T1, 0.0f, nElem);
        spmv_kernel<<<cdiv((long long)N_EDGES * F, B), B, 0, stream>>>(srcp, dstp, ew, h1, T1, F, ld);
        fill_kernel<<<cdiv(nElem, B), B, 0, stream>>>(T2, 0.0f, nElem);
        spmv_kernel<<<cdiv((long long)N_EDGES * F, B), B, 0, stream>>>(srcp, dstp, ew, T1, T2, F, ld);
        cheb_comb_kernel<<<cdiv(nElem, B), B, 0, stream>>>(T2, h1, nElem);
        dim3 g(N_NODES / 16, 32 / 16);   // 625 x 2 tiles
        // ldout == HP == 32: stores full tile; padded cols produce tanh(0)=0,
        // which is exactly the zero-padding layer 3 expects in cols 30,31.
        cheb_gemm_wmma<H1_DIM, H1_DIM, 32, 32>
            <<<g, 32, 0, stream>>>(h1, T1, T2, Wp2, bp2, h2);
    }

    // ---- layer 3 : F=30, ld=32, H=1 (scalar dot) + state assembly ----
    {
        const int F = H2_DIM, ld = 32;
        const int nElem = N_NODES * ld;
        fill_kernel<<<cdiv(nElem, B), B, 0, stream>>>(T1, 0.0f, nElem);
        spmv_kernel<<<cdiv((long long)N_EDGES * F, B), B, 0, stream>>>(srcp, dstp, ew, h2, T1, F, ld);
        fill_kernel<<<cdiv(nElem, B), B, 0, stream>>>(T2, 0.0f, nElem);
        spmv_kernel<<<cdiv((long long)N_EDGES * F, B), B, 0, stream>>>(srcp, dstp, ew, T1, T2, F, ld);
        cheb_comb_kernel<<<cdiv(nElem, B), B, 0, stream>>>(T2, h2, nElem);
        layer3_state_kernel<<<cdiv(N_NODES, B), B, 0, stream>>>(h2, T1, T2, W3, b3,
                                                                vcpu, vbw, vpend, st, ld);
    }

    // ---- actor + critic heads : stream 400MB of Wa at HBM rate ----
    {
        const long long waves = (long long)ACTION_DIM + 1;   // 10000 logits + 1 value
        head_kernel<<<cdiv(waves * 32, B), B, 0, stream>>>(st, Wa, ba, Wc, bc, out);
    }
}